// MeanAggregator_1898375545049
// MI455X (gfx1250) — compile-verified
//
#include <hip/hip_runtime.h>
#include <stdint.h>

// MeanAggregator: out[b,:] = (1/K) * sum_k table[idx[b,k], :]
// B=20000, K=32, D=128, table 500000x128 fp32.
// Gather-bandwidth bound (~0.24 FLOP/byte, ~330MB @ 23.3TB/s ~= 14us floor).
// CDNA5 async global->LDS gather path (ASYNCcnt): one instruction per 512B
// embedding row, fixed 64-bit scalar base, per-row compensation folded into
// the sign-extended 32-bit vaddr, LDS row slot via the offset: immediate.

typedef __attribute__((ext_vector_type(4))) float v4f;

#define KFAN 32
#define DIM  128

// One async row gather (GVS mode: mem = saddr + sext(vaddr.i32) + offset):
//   vaddr  = rid*512 - OFF + lane*16   (true value in (-16K, 256M): sext exact)
//   mem    = tbase + rid*512 + lane*16            -> embedding row rid
//   lds    = (base + lane*16) + OFF               -> LDS row slot k
#define GATHER_ROW(K_, OFF_)                                                  \
    {                                                                         \
        uint32_t rid = (uint32_t)__builtin_amdgcn_readlane((int)myidx, K_);   \
        uint32_t so  = (rid << 9) - (uint32_t)OFF_;   /* scalar: lshl+add */  \
        uint32_t vo  = so + voff;                     /* v_add sgpr+vgpr  */  \
        asm volatile("global_load_async_to_lds_b128 %0, %1, %2 offset:" #OFF_ \
                     :: "v"(lds_lane), "v"(vo), "s"(tbase)                    \
                     : "memory");                                             \
    }

__global__ __launch_bounds__(32) void
mean_agg_async_kernel(const float* __restrict__ table,
                      const int*   __restrict__ neigh,
                      float*       __restrict__ out)
{
    // 32 gathered rows of 512B each = 16 KB staging buffer
    __shared__ float smem[KFAN * DIM];

    const int node = blockIdx.x;
    const int lane = threadIdx.x;      // 0..31 (wave32)

    // lane k holds neighbor index k of this node (coalesced 128B read)
    const uint32_t myidx =
        (uint32_t)__builtin_nontemporal_load(&neigh[(size_t)node * KFAN + lane]);

    // LDS byte address: flat-address low 32 bits == LDS offset (ISA 10.2)
    const uint32_t lds_base = (uint32_t)(uintptr_t)(&smem[0]);
    const uint32_t lds_lane = lds_base + (uint32_t)lane * 16u;   // vdst VGPR
    const uint32_t voff     = (uint32_t)lane * 16u;              // lane chunk
    const unsigned long long tbase = (unsigned long long)(uintptr_t)table;

    // Issue all 32 row gathers (16KB in flight per wave, 1 ASYNCcnt each).
    GATHER_ROW(0, 0)       GATHER_ROW(1, 512)    GATHER_ROW(2, 1024)
    GATHER_ROW(3, 1536)    GATHER_ROW(4, 2048)   GATHER_ROW(5, 2560)
    GATHER_ROW(6, 3072)    GATHER_ROW(7, 3584)   GATHER_ROW(8, 4096)
    GATHER_ROW(9, 4608)    GATHER_ROW(10, 5120)  GATHER_ROW(11, 5632)
    GATHER_ROW(12, 6144)   GATHER_ROW(13, 6656)  GATHER_ROW(14, 7168)
    GATHER_ROW(15, 7680)   GATHER_ROW(16, 8192)  GATHER_ROW(17, 8704)
    GATHER_ROW(18, 9216)   GATHER_ROW(19, 9728)  GATHER_ROW(20, 10240)
    GATHER_ROW(21, 10752)  GATHER_ROW(22, 11264) GATHER_ROW(23, 11776)
    GATHER_ROW(24, 12288)  GATHER_ROW(25, 12800) GATHER_ROW(26, 13312)
    GATHER_ROW(27, 13824)  GATHER_ROW(28, 14336) GATHER_ROW(29, 14848)
    GATHER_ROW(30, 15360)  GATHER_ROW(31, 15872)

    v4f acc = {0.0f, 0.0f, 0.0f, 0.0f};
    const v4f* sm = (const v4f*)(&smem[0]);

    // Async loads retire in order: draining ASYNCcnt to 24/16/8/0 guarantees
    // rows [0,8)/[8,16)/[16,24)/[24,32) have landed in LDS. Accumulate each
    // group while the remaining rows are still in flight.
    auto acc8 = [&](int base) {
#pragma unroll
        for (int k = 0; k < 8; ++k)
            acc += sm[(size_t)(base + k) * (DIM / 4) + lane];
    };

    asm volatile("s_wait_asynccnt 24" ::: "memory");
    acc8(0);
    asm volatile("s_wait_asynccnt 16" ::: "memory");
    acc8(8);
    asm volatile("s_wait_asynccnt 8"  ::: "memory");
    acc8(16);
    asm volatile("s_wait_asynccnt 0"  ::: "memory");
    acc8(24);

    acc *= (1.0f / (float)KFAN);

    // Streaming write-once output: non-temporal B128 store, keep L2 for the
    // embedding table (192MB L2 vs 256MB table -> row-reuse hits matter).
    v4f* op = (v4f*)(out + (size_t)node * DIM) + lane;
    __builtin_nontemporal_store(acc, op);
}

extern "C" void kernel_launch(void* const* d_in, const int* in_sizes, int n_in,
                              void* d_out, int out_size, void* d_ws, size_t ws_size,
                              hipStream_t stream)
{
    const float* table = (const float*)d_in[0];   // [V, 128] fp32
    const int*   neigh = (const int*)d_in[1];     // [B, 32] int32
    float*       out   = (float*)d_out;           // [B, 128] fp32

    const int B = in_sizes[1] / KFAN;             // 20000
    mean_agg_async_kernel<<<B, 32, 0, stream>>>(table, neigh, out);
}